// transformerblock_42984032698879
// MI455X (gfx1250) — compile-verified
//
#include <hip/hip_runtime.h>
#include <hip/hip_bf16.h>
#include <math.h>

// Graph TransformerConv stack for MI455X (gfx1250, wave32).
// GEMMs use V_WMMA_F32_16X16X4_F32 (full fp32 path — GEMM time is negligible
// vs the L2-resident edge scatter phase, so we keep reference-exact precision).

typedef __attribute__((ext_vector_type(2))) float v2f;
typedef __attribute__((ext_vector_type(8))) float v8f;

#define DD   128
#define NH   4
#define HDIM 32

// ---- monotonic float <-> uint encoding for atomicMax on floats -------------
__device__ __forceinline__ unsigned ordEnc(float f) {
    unsigned u = __float_as_uint(f);
    return (u & 0x80000000u) ? ~u : (u | 0x80000000u);
}
__device__ __forceinline__ float ordDec(unsigned u) {
    unsigned v = (u & 0x80000000u) ? (u & 0x7FFFFFFFu) : ~u;
    return __uint_as_float(v);
}

__device__ __forceinline__ float geluf(float y) {
    return 0.5f * y * (1.0f + erff(y * 0.70710678118654752f));
}

// ---- init amax/denom -------------------------------------------------------
__global__ void fill_zero_kernel(unsigned* __restrict__ amax,
                                 float* __restrict__ denom, int count) {
    int t = blockIdx.x * blockDim.x + threadIdx.x;
    if (t < count) { amax[t] = 0u; denom[t] = 0.0f; }
}

// ---- WMMA GEMM: O = X @ W + b, three weight sets per launch ----------------
// block = 256 threads (8 waves). blockIdx.x = 16-row tile of X,
// blockIdx.y selects which of up to 3 (W,b,O) sets this block computes.
// The X tile is staged once in LDS (stride 132 -> conflict-free column reads).
__global__ __launch_bounds__(256) void gemm_wmma_kernel(
    const float* __restrict__ X,
    const float* __restrict__ W0, const float* __restrict__ W1, const float* __restrict__ W2,
    const float* __restrict__ B0, const float* __restrict__ B1, const float* __restrict__ B2,
    float* __restrict__ O0, float* __restrict__ O1, float* __restrict__ O2,
    int n)
{
    __shared__ float lds[16 * 132];
    const int row0 = blockIdx.x * 16;

    for (int i = threadIdx.x; i < 16 * DD; i += 256) {
        int r = i >> 7, c = i & 127;
        int gr = row0 + r; if (gr >= n) gr = n - 1;   // clamp (N % 16 == 0 anyway)
        lds[r * 132 + c] = X[(size_t)gr * DD + c];
    }
    __syncthreads();

    const float* W; const float* B; float* O;
    if      (blockIdx.y == 0) { W = W0; B = B0; O = O0; }
    else if (blockIdx.y == 1) { W = W1; B = B1; O = O1; }
    else                      { W = W2; B = B2; O = O2; }

    const int wave = threadIdx.x >> 5;
    const int lane = threadIdx.x & 31;
    const int col0 = wave * 16;
    const int nCol = col0 + (lane & 15);
    const int kHi  = (lane >> 4) << 1;     // lanes 0-15 -> K{0,1}; 16-31 -> K{2,3}
    const int rowA = lane & 15;

    // C/D layout: VGPR r, lanes 0-15 -> M=r, lanes 16-31 -> M=8+r; N = lane%16.
    // Bias varies only with N -> same value in all 8 accumulator VGPRs.
    const float bias = B[nCol];
    v8f c;
#pragma unroll
    for (int r = 0; r < 8; ++r) c[r] = bias;

#pragma unroll 4
    for (int k0 = 0; k0 < DD; k0 += 4) {
        v2f a, b;
        a.x = lds[rowA * 132 + k0 + kHi];
        a.y = lds[rowA * 132 + k0 + kHi + 1];
        b.x = W[(size_t)(k0 + kHi)     * DD + nCol];
        b.y = W[(size_t)(k0 + kHi + 1) * DD + nCol];
        // D = A(16x4) * B(4x16) + C, fp32 throughout
        c = __builtin_amdgcn_wmma_f32_16x16x4_f32(false, a, false, b,
                                                  (short)0, c, false, false);
    }

    const int mBase = (lane >> 4) << 3;    // 0 or 8
    if (row0 + 16 <= n) {
        // full tile: wave-uniform, unconditional stores (clause-friendly)
        float* Obase = O + (size_t)(row0 + mBase) * DD + nCol;
#pragma unroll
        for (int r = 0; r < 8; ++r) Obase[(size_t)r * DD] = c[r];
    } else {
#pragma unroll
        for (int r = 0; r < 8; ++r) {
            int row = row0 + mBase + r;
            if (row < n) O[(size_t)row * DD + nCol] = c[r];
        }
    }
}

// ---- edge phase 1: alpha = (q[dst] . k[src]) / sqrt(32); segment max -------
__global__ __launch_bounds__(256) void edge_alpha_kernel(
    const float* __restrict__ q, const float* __restrict__ k,
    const int* __restrict__ ei, float* __restrict__ alpha,
    unsigned* __restrict__ amax, int e)
{
    int t = blockIdx.x * blockDim.x + threadIdx.x;
    if (t >= e * NH) return;
    int ed = t >> 2, h = t & 3;
    int src = ei[ed], dst = ei[e + ed];
    const float4* qp = (const float4*)(q + (size_t)dst * DD + h * HDIM);
    const float4* kp = (const float4*)(k + (size_t)src * DD + h * HDIM);
    float s = 0.0f;
#pragma unroll
    for (int i = 0; i < HDIM / 4; ++i) {
        float4 a = qp[i], b = kp[i];
        s += a.x * b.x + a.y * b.y + a.z * b.z + a.w * b.w;
    }
    s *= 0.17677669529663687f;   // 1/sqrt(32)
    alpha[t] = s;
    atomicMax(&amax[(size_t)dst * NH + h], ordEnc(s));
}

// ---- edge phase 2: ex = exp(alpha - amax[dst]); segment sum ----------------
__global__ __launch_bounds__(256) void edge_softmax_kernel(
    float* __restrict__ alpha, const unsigned* __restrict__ amax,
    float* __restrict__ denom, const int* __restrict__ ei, int e)
{
    int t = blockIdx.x * blockDim.x + threadIdx.x;
    if (t >= e * NH) return;
    int ed = t >> 2, h = t & 3;
    int dst = ei[e + ed];
    float am = ordDec(amax[(size_t)dst * NH + h]);
    float ex = __expf(alpha[t] - am);
    alpha[t] = ex;                       // reuse buffer: now holds ex
    atomicAdd(&denom[(size_t)dst * NH + h], ex);
}

// ---- edge phase 3: out[dst] += v[src] * ex/(denom[dst]+eps) ----------------
// one thread per (edge, 4-float chunk): b128 gather of v, 4 f32 atomics out.
__global__ __launch_bounds__(256) void edge_msg_kernel(
    const float* __restrict__ v, const float* __restrict__ ex,
    const float* __restrict__ denom, const int* __restrict__ ei,
    float* __restrict__ out, int e)
{
    int t = blockIdx.x * blockDim.x + threadIdx.x;
    if (t >= e * (DD / 4)) return;
    int ed = t >> 5, d4 = t & 31, h = d4 >> 3;     // 32 chunks/edge, 8 per head
    int src = ei[ed], dst = ei[e + ed];
    float attn = ex[(size_t)ed * NH + h] /
                 (denom[(size_t)dst * NH + h] + 1e-16f);
    const float4 vv = *(const float4*)(v + (size_t)src * DD + d4 * 4);
    float* o = out + (size_t)dst * DD + d4 * 4;
    atomicAdd(o + 0, vv.x * attn);
    atomicAdd(o + 1, vv.y * attn);
    atomicAdd(o + 2, vv.z * attn);
    atomicAdd(o + 3, vv.w * attn);
}

// ---- node phase: out = gelu(layernorm(conv)) + res; one wave32 per node ----
__global__ __launch_bounds__(256) void ln_gelu_res_kernel(
    const float* __restrict__ conv, const float* __restrict__ res,
    const float* __restrict__ g, const float* __restrict__ b,
    float* __restrict__ out, int n)
{
    int node = blockIdx.x * 8 + (threadIdx.x >> 5);
    if (node >= n) return;                    // uniform across the wave
    int lane = threadIdx.x & 31;
    const float4 xv = *(const float4*)(conv + (size_t)node * DD + lane * 4);

    float s = xv.x + xv.y + xv.z + xv.w;
#pragma unroll
    for (int m = 16; m >= 1; m >>= 1) s += __shfl_xor(s, m, 32);
    const float mu = s * (1.0f / 128.0f);

    float d0 = xv.x - mu, d1 = xv.y - mu, d2 = xv.z - mu, d3 = xv.w - mu;
    float vs = d0 * d0 + d1 * d1 + d2 * d2 + d3 * d3;
#pragma unroll
    for (int m = 16; m >= 1; m >>= 1) vs += __shfl_xor(vs, m, 32);
    const float rstd = rsqrtf(vs * (1.0f / 128.0f) + 1e-5f);

    const float4 gv = *(const float4*)(g + lane * 4);
    const float4 bv = *(const float4*)(b + lane * 4);
    const float4 rv = *(const float4*)(res + (size_t)node * DD + lane * 4);
    float4 o;
    o.x = geluf(d0 * rstd * gv.x + bv.x) + rv.x;
    o.y = geluf(d1 * rstd * gv.y + bv.y) + rv.y;
    o.z = geluf(d2 * rstd * gv.z + bv.z) + rv.z;
    o.w = geluf(d3 * rstd * gv.w + bv.w) + rv.w;
    *(float4*)(out + (size_t)node * DD + lane * 4) = o;
}

// ---- final layer: elementwise GELU ----------------------------------------
__global__ __launch_bounds__(256) void gelu_kernel(
    const float* __restrict__ in, float* __restrict__ out, int count)
{
    int t = blockIdx.x * blockDim.x + threadIdx.x;
    if (t < count) out[t] = geluf(in[t]);
}

extern "C" void kernel_launch(void* const* d_in, const int* in_sizes, int n_in,
                              void* d_out, int out_size, void* d_ws, size_t ws_size,
                              hipStream_t stream)
{
    const float* x    = (const float*)d_in[0];
    const int*   ei   = (const int*)  d_in[1];
    const float* Wq   = (const float*)d_in[2];
    const float* bq   = (const float*)d_in[3];
    const float* Wk   = (const float*)d_in[4];
    const float* bk   = (const float*)d_in[5];
    const float* Wv   = (const float*)d_in[6];
    const float* bv   = (const float*)d_in[7];
    const float* Wsp  = (const float*)d_in[8];
    const float* bs   = (const float*)d_in[9];
    const float* ln_g = (const float*)d_in[10];
    const float* ln_b = (const float*)d_in[11];
    float* out = (float*)d_out;

    const int n = in_sizes[0] / DD;       // 50000
    const int e = in_sizes[1] / 2;        // 800000

    // workspace carve-up (fp32 units)
    float* ws     = (float*)d_ws;
    float* hbuf   = ws;                    ws += (size_t)n * DD;
    float* qbuf   = ws;                    ws += (size_t)n * DD;   // aliased as aggr
    float* kbuf   = ws;                    ws += (size_t)n * DD;
    float* vbuf   = ws;                    ws += (size_t)n * DD;
    float* exbuf  = ws;                    ws += (size_t)e * NH;
    unsigned* amax = (unsigned*)ws;        ws += (size_t)n * NH;
    float* denom  = ws;                    ws += (size_t)n * NH;
    float* aggr   = qbuf;                  // q dead after edge_alpha

    const dim3 blk(256);
    const int rowTiles  = (n + 15) / 16;
    const int ehThreads = e * NH;
    const int ecThreads = e * (DD / 4);

    const float* hin = x;
    for (int layer = 0; layer < 5; ++layer) {
        const size_t wOff = (size_t)layer * DD * DD;
        const size_t bOff = (size_t)layer * DD;

        fill_zero_kernel<<<(n * NH + 255) / 256, blk, 0, stream>>>(amax, denom, n * NH);

        // q, k, v projections (WMMA)
        gemm_wmma_kernel<<<dim3(rowTiles, 3), blk, 0, stream>>>(
            hin, Wq + wOff, Wk + wOff, Wv + wOff,
            bq + bOff, bk + bOff, bv + bOff,
            qbuf, kbuf, vbuf, n);

        edge_alpha_kernel<<<(ehThreads + 255) / 256, blk, 0, stream>>>(
            qbuf, kbuf, ei, exbuf, amax, e);

        // s projection seeds the aggregation buffer: aggr = hin@Ws + bs
        gemm_wmma_kernel<<<dim3(rowTiles, 1), blk, 0, stream>>>(
            hin, Wsp + wOff, Wsp + wOff, Wsp + wOff,
            bs + bOff, bs + bOff, bs + bOff,
            aggr, aggr, aggr, n);

        edge_softmax_kernel<<<(ehThreads + 255) / 256, blk, 0, stream>>>(
            exbuf, amax, denom, ei, e);

        edge_msg_kernel<<<(ecThreads + 255) / 256, blk, 0, stream>>>(
            vbuf, exbuf, denom, ei, aggr, e);

        if (layer < 4) {
            ln_gelu_res_kernel<<<(n + 7) / 8, blk, 0, stream>>>(
                aggr, hin, ln_g + (size_t)layer * DD, ln_b + (size_t)layer * DD,
                hbuf, n);
            hin = hbuf;   // layers 1-3 update hbuf in place (elementwise-safe)
        } else {
            gelu_kernel<<<(n * DD + 255) / 256, blk, 0, stream>>>(aggr, out, n * DD);
        }
    }
}